// GraphSAGE_12558484373968
// MI455X (gfx1250) — compile-verified
//
#include <hip/hip_runtime.h>
#include <hip/hip_bf16.h>

// GraphSAGE 2-layer, fp32 end-to-end on gfx1250.
// Matmuls: V_WMMA_F32_16X16X4_F32 (exact fp32 WMMA). Workload is
// aggregation/L2-bandwidth bound, so fp32 matrix math costs nothing extra.
// GEMM is tiled 32x64 per wave: 8 accumulators, each B load feeds 2 row-tiles
// (halves W streaming traffic), each A load feeds 4 column-tiles. B operands
// use distinct registers so loads pipeline across the 8-WMMA burst.
// Mean-normalization is folded into the A operand (row-uniform scale).

typedef __attribute__((ext_vector_type(2))) float v2f;
typedef __attribute__((ext_vector_type(8))) float v8f;

constexpr int NNODES = 50000;
constexpr int NEDGES = 800000;
constexpr int DIN  = 256;
constexpr int DHID = 256;
constexpr int DOUT = 128;

#define WMMA_F32(a, b, c) \
    __builtin_amdgcn_wmma_f32_16x16x4_f32(false, (a), false, (b), (short)0, (c), false, false)

// ---------------- zero fill (float4) ----------------
__global__ void zero_f4(float4* __restrict__ p, int n4) {
    int i = blockIdx.x * blockDim.x + threadIdx.x;
    if (i < n4) p[i] = make_float4(0.f, 0.f, 0.f, 0.f);
}

// ---------------- edge scatter: agg[dst] += feat[src]; cnt[dst] += 1 -------
// 64 threads per edge, 4 floats each (float4 gather, 4x global_atomic_add_f32)
__global__ void scatter_add(const int* __restrict__ src, const int* __restrict__ dst,
                            const float* __restrict__ feat, float* __restrict__ agg,
                            float* __restrict__ cnt, int nE) {
    int t = blockIdx.x * blockDim.x + threadIdx.x;
    int e = t >> 6;
    if (e >= nE) return;
    int f = (t & 63) << 2;                 // feature offset 0..252
    int s = src[e];
    int d = dst[e];
    float4 v = *(const float4*)(feat + (size_t)s * 256 + f);
    float* ap = agg + (size_t)d * 256 + f;
    atomicAdd(ap + 0, v.x);
    atomicAdd(ap + 1, v.y);
    atomicAdd(ap + 2, v.z);
    atomicAdd(ap + 3, v.w);
    if (f == 0) atomicAdd(cnt + d, 1.0f);
}

// ---------------- fused SAGE layer GEMM ------------------------------------
// out[M,N] = act( (Aagg/max(cnt,1)) @ Wl + Aself @ Wr + bias )
// 32x64 tile per wave (2 row-tiles x 4 col-tiles), fp32 WMMA 16x16x4.
__global__ __launch_bounds__(256) void sage_gemm_wmma(
        const float* __restrict__ Aagg,  const float* __restrict__ Aself,
        const float* __restrict__ cnt,
        const float* __restrict__ Wl,    const float* __restrict__ Wr,
        const float* __restrict__ bias,  float* __restrict__ out,
        int M, int K, int N, int doRelu) {
    int wave    = blockIdx.x * (blockDim.x >> 5) + (threadIdx.x >> 5);
    int nG      = N >> 6;                 // 64-column groups
    int nTilesM = M >> 4;                 // 16-row tiles
    int nPairsM = (nTilesM + 1) >> 1;     // 32-row pairs (last may be partial)
    int nTiles  = nPairsM * nG;
    if (wave >= nTiles) return;           // whole-wave uniform: EXEC stays all-1s

    int pairM = wave / nG;
    int g     = wave % nG;
    int lane  = threadIdx.x & 31;
    int l15   = lane & 15;
    int kHi   = lane >> 4;                // 0: K pair {k,k+1}, 1: {k+2,k+3}

    int tileM0 = pairM * 2;
    int tileM1 = tileM0 + 1;
    bool hasM1 = (tileM1 < nTilesM);      // wave-uniform
    int tileM1e = hasM1 ? tileM1 : tileM0;  // alias row-tile 0 when absent

    int row0 = tileM0  * 16 + l15;        // A rows (same for both half-waves)
    int row1 = tileM1e * 16 + l15;
    int col  = g * 64 + l15;              // first of 4 column sub-tiles

    float c0 = cnt[row0], c1 = cnt[row1];
    float s0 = c0 > 1.0f ? 1.0f / c0 : 1.0f;
    float s1 = c1 > 1.0f ? 1.0f / c1 : 1.0f;

    float bv0 = bias[col];
    float bv1 = bias[col + 16];
    float bv2 = bias[col + 32];
    float bv3 = bias[col + 48];
    v8f acc00 = {bv0, bv0, bv0, bv0, bv0, bv0, bv0, bv0};
    v8f acc01 = {bv1, bv1, bv1, bv1, bv1, bv1, bv1, bv1};
    v8f acc02 = {bv2, bv2, bv2, bv2, bv2, bv2, bv2, bv2};
    v8f acc03 = {bv3, bv3, bv3, bv3, bv3, bv3, bv3, bv3};
    v8f acc10 = acc00, acc11 = acc01, acc12 = acc02, acc13 = acc03;

    // ---- pass 1: mean @ Wl (scale fused into A) ----
    {
        const float* a0p = Aagg + (size_t)row0 * K + 2 * kHi;
        const float* a1p = Aagg + (size_t)row1 * K + 2 * kHi;
        const float* b0p = Wl + (size_t)(2 * kHi) * N + col;
        #pragma unroll 4
        for (int k = 0; k < K; k += 4) {
            const float* bk = b0p + (size_t)k * N;
            v2f a0 = *(const v2f*)(a0p + k);        // A: K = k+2*kHi, k+2*kHi+1
            v2f a1 = *(const v2f*)(a1p + k);
            v2f bA, bB, bC, bD;                     // distinct regs -> pipelined
            bA.x = bk[0];  bA.y = bk[(size_t)N];
            bB.x = bk[16]; bB.y = bk[(size_t)N + 16];
            bC.x = bk[32]; bC.y = bk[(size_t)N + 32];
            bD.x = bk[48]; bD.y = bk[(size_t)N + 48];
            a0.x *= s0; a0.y *= s0;
            a1.x *= s1; a1.y *= s1;
            acc00 = WMMA_F32(a0, bA, acc00);
            acc10 = WMMA_F32(a1, bA, acc10);
            acc01 = WMMA_F32(a0, bB, acc01);
            acc11 = WMMA_F32(a1, bB, acc11);
            acc02 = WMMA_F32(a0, bC, acc02);
            acc12 = WMMA_F32(a1, bC, acc12);
            acc03 = WMMA_F32(a0, bD, acc03);
            acc13 = WMMA_F32(a1, bD, acc13);
        }
    }
    // ---- pass 2: self @ Wr ----
    {
        const float* a0p = Aself + (size_t)row0 * K + 2 * kHi;
        const float* a1p = Aself + (size_t)row1 * K + 2 * kHi;
        const float* b0p = Wr + (size_t)(2 * kHi) * N + col;
        #pragma unroll 4
        for (int k = 0; k < K; k += 4) {
            const float* bk = b0p + (size_t)k * N;
            v2f a0 = *(const v2f*)(a0p + k);
            v2f a1 = *(const v2f*)(a1p + k);
            v2f bA, bB, bC, bD;
            bA.x = bk[0];  bA.y = bk[(size_t)N];
            bB.x = bk[16]; bB.y = bk[(size_t)N + 16];
            bC.x = bk[32]; bC.y = bk[(size_t)N + 32];
            bD.x = bk[48]; bD.y = bk[(size_t)N + 48];
            acc00 = WMMA_F32(a0, bA, acc00);
            acc10 = WMMA_F32(a1, bA, acc10);
            acc01 = WMMA_F32(a0, bB, acc01);
            acc11 = WMMA_F32(a1, bB, acc11);
            acc02 = WMMA_F32(a0, bC, acc02);
            acc12 = WMMA_F32(a1, bC, acc12);
            acc03 = WMMA_F32(a0, bD, acc03);
            acc13 = WMMA_F32(a1, bD, acc13);
        }
    }

    if (doRelu) {
        #pragma unroll
        for (int r = 0; r < 8; ++r) {
            acc00[r] = fmaxf(acc00[r], 0.0f); acc01[r] = fmaxf(acc01[r], 0.0f);
            acc02[r] = fmaxf(acc02[r], 0.0f); acc03[r] = fmaxf(acc03[r], 0.0f);
            acc10[r] = fmaxf(acc10[r], 0.0f); acc11[r] = fmaxf(acc11[r], 0.0f);
            acc12[r] = fmaxf(acc12[r], 0.0f); acc13[r] = fmaxf(acc13[r], 0.0f);
        }
    }

    // D layout: VGPR r -> row r (lanes 0-15) / row r+8 (lanes 16-31)
    {
        float* orow = out + (size_t)(tileM0 * 16 + 8 * kHi) * N + col;
        #pragma unroll
        for (int r = 0; r < 8; ++r) {
            orow[(size_t)r * N +  0] = acc00[r];
            orow[(size_t)r * N + 16] = acc01[r];
            orow[(size_t)r * N + 32] = acc02[r];
            orow[(size_t)r * N + 48] = acc03[r];
        }
    }
    if (hasM1) {                           // wave-uniform branch
        float* orow = out + (size_t)(tileM1 * 16 + 8 * kHi) * N + col;
        #pragma unroll
        for (int r = 0; r < 8; ++r) {
            orow[(size_t)r * N +  0] = acc10[r];
            orow[(size_t)r * N + 16] = acc11[r];
            orow[(size_t)r * N + 32] = acc12[r];
            orow[(size_t)r * N + 48] = acc13[r];
        }
    }
}

extern "C" void kernel_launch(void* const* d_in, const int* in_sizes, int n_in,
                              void* d_out, int out_size, void* d_ws, size_t ws_size,
                              hipStream_t stream) {
    const float* x   = (const float*)d_in[0];
    const int*   ei  = (const int*)d_in[1];
    const float* W1l = (const float*)d_in[2];
    const float* b1  = (const float*)d_in[3];
    const float* W1r = (const float*)d_in[4];
    const float* W2l = (const float*)d_in[5];
    const float* b2  = (const float*)d_in[6];
    const float* W2r = (const float*)d_in[7];
    float* out = (float*)d_out;

    const int* src = ei;            // edge_index[0]
    const int* dst = ei + NEDGES;   // edge_index[1]

    // workspace: agg[50000*256] | h[50000*256] | cnt[50000]  (~102.6 MB)
    float* agg = (float*)d_ws;
    float* h   = agg + (size_t)NNODES * DHID;
    float* cnt = h   + (size_t)NNODES * DHID;

    const int nAgg  = NNODES * DIN;                  // 12.8M floats
    const int nAgg4 = nAgg / 4;
    const int nCnt4 = NNODES / 4;
    const int tSc   = NEDGES * 64;                   // 51.2M scatter threads

    const int nPairsM = (NNODES / 16 + 1) / 2;       // 1563 (last pair partial)
    const int tiles1  = nPairsM * (DHID / 64);       // 6252 waves (32x64 tiles)
    const int tiles2  = nPairsM * (DOUT / 64);       // 3126 waves

    // ---------------- layer 1 ----------------
    zero_f4<<<(nAgg4 + 255) / 256, 256, 0, stream>>>((float4*)agg, nAgg4);
    zero_f4<<<(nCnt4 + 255) / 256, 256, 0, stream>>>((float4*)cnt, nCnt4);
    scatter_add<<<(tSc + 255) / 256, 256, 0, stream>>>(src, dst, x, agg, cnt, NEDGES);
    sage_gemm_wmma<<<(tiles1 + 7) / 8, 256, 0, stream>>>(
        agg, x, cnt, W1l, W1r, b1, h, NNODES, DIN, DHID, /*relu=*/1);

    // ---------------- layer 2 ----------------
    zero_f4<<<(nAgg4 + 255) / 256, 256, 0, stream>>>((float4*)agg, nAgg4);
    zero_f4<<<(nCnt4 + 255) / 256, 256, 0, stream>>>((float4*)cnt, nCnt4);
    scatter_add<<<(tSc + 255) / 256, 256, 0, stream>>>(src, dst, h, agg, cnt, NEDGES);
    sage_gemm_wmma<<<(tiles2 + 7) / 8, 256, 0, stream>>>(
        agg, h, cnt, W2l, W2r, b2, out, NNODES, DHID, DOUT, /*relu=*/0);
}